// OHEM_76768245449349
// MI455X (gfx1250) — compile-verified
//
#include <hip/hip_runtime.h>
#include <math.h>

// ---------------- problem constants ----------------
#define ROWS 64
#define HW   409600            // 640*640
#define THRF 0.01f

// ---------------- launch geometry ------------------
#define BPR      100                    // blocks per row
#define EPB      (HW / BPR)             // 4096 elements per block
#define THREADS  256
#define F4PT     (EPB / 4 / THREADS)    // 4 float4 per thread

#define NBIN1 2048                      // bits[30:20]
#define NBIN2 4096                      // bits[19:8]

// ---------------- workspace layout (bytes) ---------
#define OFF_HIST1   0u                                  // u32 [ROWS][NBIN1]   512 KB
#define OFF_POS     (OFF_HIST1 + ROWS*NBIN1*4u)         // u32 [ROWS]
#define OFF_SELB    (OFF_POS   + ROWS*4u)               // i32 [ROWS]
#define OFF_SELK2   (OFF_SELB  + ROWS*4u)               // u32 [ROWS]
#define OFF_SELNHE  (OFF_SELK2 + ROWS*4u)               // f32 [ROWS]
#define OFF_HIST2   (OFF_SELNHE+ ROWS*4u)               // u32 [ROWS][NBIN2]   1 MB
#define OFF_SUM2    (OFF_HIST2 + ROWS*NBIN2*4u)         // f32 [ROWS][NBIN2]   1 MB
#define OFF_ABOVE   (OFF_SUM2  + ROWS*NBIN2*4u)         // f64 [ROWS][BPR]     51 KB
#define WS_BYTES    (OFF_ABOVE + ROWS*BPR*8u)

// ---------------- CDNA5 async-to-LDS staging -------
#if defined(__AMDGCN__) && \
    __has_builtin(__builtin_amdgcn_global_load_async_to_lds_b128) && \
    __has_builtin(__builtin_amdgcn_s_wait_asynccnt)
#define USE_ASYNC_LDS 1
#else
#define USE_ASYNC_LDS 0
#endif

typedef int v4i __attribute__((vector_size(16)));

#if USE_ASYNC_LDS
typedef __attribute__((address_space(1))) v4i* gv4i_p;   // global (HIP __device__)
typedef __attribute__((address_space(3))) v4i* lv4i_p;   // LDS (__shared__)

__device__ __forceinline__ void async_cp16(const float4* g, float4* l) {
    v4i* gg = (v4i*)(void*)const_cast<float4*>(g);
    v4i* ll = (v4i*)(void*)l;
    __builtin_amdgcn_global_load_async_to_lds_b128((gv4i_p)gg, (lv4i_p)ll, 0, 0);
}
#endif

// Stage buffers: [buf][x/y][thread] float4 -> 2*2*256*16B = 16 KB
#define STAGE_DECL __shared__ float4 stage[2][2][THREADS]

#if USE_ASYNC_LDS
#define STAGE_ISSUE(buf, k)                                                     \
    do {                                                                        \
        const int _i = threadIdx.x + (k) * THREADS;                             \
        async_cp16(x4 + _i, &stage[(buf)][0][threadIdx.x]);                     \
        async_cp16(y4 + _i, &stage[(buf)][1][threadIdx.x]);                     \
    } while (0)
#define STAGE_WAIT_CUR()  do { __builtin_amdgcn_s_wait_asynccnt(2); \
                               __asm__ volatile("" ::: "memory"); } while (0)
#define STAGE_WAIT_ALL()  do { __builtin_amdgcn_s_wait_asynccnt(0); \
                               __asm__ volatile("" ::: "memory"); } while (0)
#else
#define STAGE_ISSUE(buf, k)                                                     \
    do {                                                                        \
        const int _i = threadIdx.x + (k) * THREADS;                             \
        stage[(buf)][0][threadIdx.x] = x4[_i];                                  \
        stage[(buf)][1][threadIdx.x] = y4[_i];                                  \
    } while (0)
#define STAGE_WAIT_CUR()  do { } while (0)
#define STAGE_WAIT_ALL()  do { } while (0)
#endif

// ======================================================================
// Pass 1: per-row coarse histogram (bits[30:20]) + positive counts
// ======================================================================
__global__ __launch_bounds__(THREADS)
void ohem_pass1(const float* __restrict__ x, const float* __restrict__ y,
                unsigned* __restrict__ hist1, unsigned* __restrict__ poscnt) {
    __shared__ unsigned lh[NBIN1];
    __shared__ unsigned lpos;
    STAGE_DECL;
    for (int i = threadIdx.x; i < NBIN1; i += THREADS) lh[i] = 0u;
    if (threadIdx.x == 0) lpos = 0u;
    __syncthreads();

    const int row = blockIdx.x / BPR;
    const int cid = blockIdx.x % BPR;
    const size_t base = (size_t)row * HW + (size_t)cid * EPB;
    const float4* x4 = (const float4*)(x + base);
    const float4* y4 = (const float4*)(y + base);

    unsigned mypos = 0u;
    STAGE_ISSUE(0, 0);
#pragma unroll
    for (int k = 0; k < F4PT; ++k) {
        const int buf = k & 1;
        if (k + 1 < F4PT) { STAGE_ISSUE(buf ^ 1, k + 1); STAGE_WAIT_CUR(); }
        else              { STAGE_WAIT_ALL(); }
        float4 a = stage[buf][0][threadIdx.x];
        float4 b = stage[buf][1][threadIdx.x];
        const float d0 = a.x - b.x, d1 = a.y - b.y, d2 = a.z - b.z, d3 = a.w - b.w;
        const float l[4] = {d0 * d0, d1 * d1, d2 * d2, d3 * d3};
#pragma unroll
        for (int j = 0; j < 4; ++j) {
            if (l[j] < THRF) { mypos++; }
            else { atomicAdd(&lh[__float_as_uint(l[j]) >> 20], 1u); }
        }
    }
    atomicAdd(&lpos, mypos);
    __syncthreads();

    unsigned* gh = hist1 + (size_t)row * NBIN1;
    for (int i = threadIdx.x; i < NBIN1; i += THREADS) {
        const unsigned c = lh[i];
        if (c) atomicAdd(&gh[i], c);
    }
    if (threadIdx.x == 0 && lpos) atomicAdd(&poscnt[row], lpos);
}

// ======================================================================
// Select 1: khe, threshold coarse bin b, remaining count k2
// ======================================================================
__global__ void ohem_select1(const unsigned* __restrict__ hist1,
                             const unsigned* __restrict__ poscnt,
                             int* __restrict__ selb, unsigned* __restrict__ selk2,
                             float* __restrict__ selnhe) {
    const int r = threadIdx.x;
    if (r >= ROWS) return;
    const unsigned npos = poscnt[r];
    const float nposf = (float)npos;
    const float nnegf = (float)(HW - npos);
    float nhe = (nnegf > 3.0f * nposf) ? (3.0f * nposf) : nnegf;
    nhe = fmaxf(nhe, 2048.0f);               // HE_RATIO * HW = 2048 exactly
    const unsigned khe = (unsigned)nhe;      // nhe is integer-valued

    const unsigned* h = hist1 + (size_t)r * NBIN1;
    unsigned cum = 0u;
    int b = -1;
    for (int bin = NBIN1 - 1; bin >= 0; --bin) {
        const unsigned c = h[bin];
        if (cum + c >= khe) { b = bin; break; }
        cum += c;
    }
    selb[r]   = b;
    selk2[r]  = khe - cum;   // still to take from bin b (or zeros if b == -1)
    selnhe[r] = nhe;
}

// ======================================================================
// Pass 2: f64 sum of bins > b, refine bin b with bits[19:8] hist+sums
// ======================================================================
__global__ __launch_bounds__(THREADS)
void ohem_pass2(const float* __restrict__ x, const float* __restrict__ y,
                const int* __restrict__ selb,
                unsigned* __restrict__ hist2, float* __restrict__ sum2,
                double* __restrict__ abovePart) {
    __shared__ unsigned h2c[NBIN2];
    __shared__ float    h2s[NBIN2];
    __shared__ double   red[THREADS];
    STAGE_DECL;
    for (int i = threadIdx.x; i < NBIN2; i += THREADS) { h2c[i] = 0u; h2s[i] = 0.0f; }
    __syncthreads();

    const int row = blockIdx.x / BPR;
    const int cid = blockIdx.x % BPR;
    const int b   = selb[row];
    const size_t base = (size_t)row * HW + (size_t)cid * EPB;
    const float4* x4 = (const float4*)(x + base);
    const float4* y4 = (const float4*)(y + base);

    double local = 0.0;
    STAGE_ISSUE(0, 0);
#pragma unroll
    for (int k = 0; k < F4PT; ++k) {
        const int buf = k & 1;
        if (k + 1 < F4PT) { STAGE_ISSUE(buf ^ 1, k + 1); STAGE_WAIT_CUR(); }
        else              { STAGE_WAIT_ALL(); }
        float4 a  = stage[buf][0][threadIdx.x];
        float4 bb = stage[buf][1][threadIdx.x];
        const float d[4] = {a.x - bb.x, a.y - bb.y, a.z - bb.z, a.w - bb.w};
#pragma unroll
        for (int j = 0; j < 4; ++j) {
            const float l = d[j] * d[j];
            if (!(l < THRF)) {
                const unsigned u = __float_as_uint(l);
                const int bin1 = (int)(u >> 20);
                if (bin1 > b) {
                    local += (double)l;
                } else if (bin1 == b) {
                    const int bin2 = (int)((u >> 8) & 0xFFFu);
                    atomicAdd(&h2c[bin2], 1u);
                    atomicAdd(&h2s[bin2], l);
                }
            }
        }
    }
    red[threadIdx.x] = local;
    __syncthreads();
    for (int s = THREADS / 2; s > 0; s >>= 1) {
        if (threadIdx.x < s) red[threadIdx.x] += red[threadIdx.x + s];
        __syncthreads();
    }
    if (threadIdx.x == 0) abovePart[(size_t)row * BPR + cid] = red[0];

    unsigned* gh = hist2 + (size_t)row * NBIN2;
    float*    gs = sum2  + (size_t)row * NBIN2;
    for (int i = threadIdx.x; i < NBIN2; i += THREADS) {
        const unsigned c = h2c[i];
        if (c) { atomicAdd(&gh[i], c); atomicAdd(&gs[i], h2s[i]); }
    }
}

// ======================================================================
// Select 2: resolve threshold, assemble per-row loss, average into d_out
// ======================================================================
__global__ void ohem_select2(const unsigned* __restrict__ hist2,
                             const float* __restrict__ sum2,
                             const int* __restrict__ selb,
                             const unsigned* __restrict__ selk2,
                             const float* __restrict__ selnhe,
                             const double* __restrict__ abovePart,
                             float* __restrict__ out) {
    const int r = threadIdx.x;
    if (r >= ROWS) return;

    double above = 0.0;
    const double* p = abovePart + (size_t)r * BPR;
    for (int i = 0; i < BPR; ++i) above += p[i];

    const unsigned k2 = selk2[r];
    const int b = selb[r];
    const unsigned* h = hist2 + (size_t)r * NBIN2;
    const float*    s = sum2  + (size_t)r * NBIN2;

    unsigned cum = 0u;
    double S2 = 0.0;
    int b2 = -1;
    for (int bin = NBIN2 - 1; bin >= 0; --bin) {
        const unsigned c = h[bin];
        if (cum + c >= k2) { b2 = bin; break; }
        cum += c;
        S2 += (double)s[bin];
    }
    double tcontrib = 0.0;
    if (b >= 0 && b2 >= 0) {
        const unsigned krem = k2 - cum;                 // copies of threshold value
        const unsigned bits = ((unsigned)b << 20) | ((unsigned)b2 << 8) | 0x80u;
        tcontrib = (double)krem * (double)__uint_as_float(bits);
    }
    const double total = above + S2 + tcontrib;
    const float rowLoss = (float)(total / (double)selnhe[r]);
    atomicAdd(out, rowLoss * (1.0f / 64.0f));
}

// ======================================================================
extern "C" void kernel_launch(void* const* d_in, const int* in_sizes, int n_in,
                              void* d_out, int out_size, void* d_ws, size_t ws_size,
                              hipStream_t stream) {
    (void)in_sizes; (void)n_in; (void)out_size; (void)ws_size;
    const float* x = (const float*)d_in[0];
    const float* y = (const float*)d_in[1];
    float* out = (float*)d_out;
    unsigned char* ws = (unsigned char*)d_ws;

    unsigned* hist1  = (unsigned*)(ws + OFF_HIST1);
    unsigned* poscnt = (unsigned*)(ws + OFF_POS);
    int*      selb   = (int*)     (ws + OFF_SELB);
    unsigned* selk2  = (unsigned*)(ws + OFF_SELK2);
    float*    selnhe = (float*)   (ws + OFF_SELNHE);
    unsigned* hist2  = (unsigned*)(ws + OFF_HIST2);
    float*    sum2   = (float*)   (ws + OFF_SUM2);
    double*   above  = (double*)  (ws + OFF_ABOVE);

    (void)hipMemsetAsync(ws, 0, WS_BYTES, stream);
    (void)hipMemsetAsync(out, 0, sizeof(float), stream);

    ohem_pass1<<<ROWS * BPR, THREADS, 0, stream>>>(x, y, hist1, poscnt);
    ohem_select1<<<1, ROWS, 0, stream>>>(hist1, poscnt, selb, selk2, selnhe);
    ohem_pass2<<<ROWS * BPR, THREADS, 0, stream>>>(x, y, selb, hist2, sum2, above);
    ohem_select2<<<1, ROWS, 0, stream>>>(hist2, sum2, selb, selk2, selnhe, above, out);
}